// BasicGNNLayer_79070347919847
// MI455X (gfx1250) — compile-verified
//
#include <hip/hip_runtime.h>

// BasicGNNLayer: out = features + segment_sum(features[col], row) / clip(deg,1)
// N = 100000 nodes, E = 1600000 edges, D = 64 f32 features.
//
// Strategy (L2-resident scatter/gather):
//   K1 (hot): 16 lanes/edge, float4 gather of source row (global_load_b128),
//       hardware fp32 atomics into d_out (global_atomic_add_f32),
//       u32 atomics for degree, global_prefetch_b8 for next iteration's gather
//   K0: zero d_out (accumulator) and deg counters in d_ws
//   K2: out = features + out * (1 / max(deg,1)), float4 per thread
//
// Edge kernel is placed FIRST in the file so the disasm snippet shows its
// atomic / prefetch instruction selection.

#define GNN_D_FEAT 64
#define GNN_D_VEC4 16  // 64 floats = 16 float4 per row

__global__ void gnn_edge_scatter_kernel(const float4* __restrict__ feat4,
                                        const int* __restrict__ row,
                                        const int* __restrict__ col,
                                        float* __restrict__ agg,
                                        unsigned int* __restrict__ deg,
                                        int n_edges) {
  unsigned int gid = blockIdx.x * blockDim.x + threadIdx.x;
  const int q = (int)(gid & 15u);          // float4 slot within the 64-float row
  int e = (int)(gid >> 4);                 // edge index
  const int estride = (int)((gridDim.x * blockDim.x) >> 4);

  for (; e < n_edges; e += estride) {
    const int c = col[e];
    const int r = row[e];

    // Prefetch the NEXT iteration's random gather target (gfx1250
    // global_prefetch_b8). The index load col[e+estride] is a cheap,
    // cache-friendly sequential read; the feature-row fetch is the
    // long-latency random access we want in flight early.
    if (e + estride < n_edges) {
      const int cn = col[e + estride];
      __builtin_prefetch(&feat4[(size_t)cn * GNN_D_VEC4 + q], 0, 0);
    }

    // Coalesced 256B gather across 16 lanes: one b128 load per lane.
    const float4 v = feat4[(size_t)c * GNN_D_VEC4 + q];

    // Segment-sum via native L2 fp32 atomics (non-returning).
    float* dst = agg + ((size_t)r * GNN_D_FEAT + (size_t)q * 4);
    unsafeAtomicAdd(dst + 0, v.x);
    unsafeAtomicAdd(dst + 1, v.y);
    unsafeAtomicAdd(dst + 2, v.z);
    unsafeAtomicAdd(dst + 3, v.w);

    // One lane per edge bumps the degree counter.
    if (q == 0) atomicAdd(deg + r, 1u);
  }
}

__global__ void gnn_zero_kernel(float* __restrict__ out,
                                unsigned int* __restrict__ deg,
                                size_t n_out, int n_nodes) {
  size_t t = (size_t)blockIdx.x * blockDim.x + threadIdx.x;
  if (t < n_out) out[t] = 0.0f;
  if (t < (size_t)n_nodes) deg[t] = 0u;
}

__global__ void gnn_finalize_kernel(const float4* __restrict__ feat4,
                                    float4* __restrict__ out4,
                                    const unsigned int* __restrict__ deg,
                                    int n_nodes) {
  const int t = blockIdx.x * blockDim.x + threadIdx.x;
  const int total = n_nodes * GNN_D_VEC4;
  if (t >= total) return;

  const unsigned int d = deg[t >> 4];
  const float inv = 1.0f / (float)(d > 0u ? d : 1u);

  const float4 a = out4[t];
  const float4 f = feat4[t];
  float4 r;
  r.x = f.x + a.x * inv;
  r.y = f.y + a.y * inv;
  r.z = f.z + a.z * inv;
  r.w = f.w + a.w * inv;
  out4[t] = r;
}

extern "C" void kernel_launch(void* const* d_in, const int* in_sizes, int n_in,
                              void* d_out, int out_size, void* d_ws, size_t ws_size,
                              hipStream_t stream) {
  const float* feat = (const float*)d_in[0];   // [N, 64] f32
  const int* row = (const int*)d_in[1];        // [E] i32
  const int* col = (const int*)d_in[2];        // [E] i32
  float* out = (float*)d_out;                  // [N, 64] f32

  const int n_nodes = in_sizes[0] / GNN_D_FEAT;
  const int n_edges = in_sizes[1];
  unsigned int* deg = (unsigned int*)d_ws;     // N u32 counters (400 KB)

  // K0: zero the accumulator (d_out) and degree counters.
  const size_t n_out = (size_t)n_nodes * GNN_D_FEAT;
  const int zblocks = (int)((n_out + 255) / 256);
  gnn_zero_kernel<<<zblocks, 256, 0, stream>>>(out, deg, n_out, n_nodes);

  // K1: edge scatter. Grid-stride over E*16 lane-slots; fixed 8192 blocks
  // so each thread iterates ~12x and the prefetch pipeline pays off.
  gnn_edge_scatter_kernel<<<8192, 256, 0, stream>>>(
      (const float4*)feat, row, col, out, deg, n_edges);

  // K2: normalize + residual.
  const int total4 = n_nodes * GNN_D_VEC4;
  gnn_finalize_kernel<<<(total4 + 255) / 256, 256, 0, stream>>>(
      (const float4*)feat, (float4*)out, deg, n_nodes);
}